// Model_61108794687954
// MI455X (gfx1250) — compile-verified
//
#include <hip/hip_runtime.h>
#include <math.h>

#define N_SAMPLES 32768
#define N_ATOMS   2048
#define ATOM_SIZE 1024
#define K_SPARSE  64
#define MD        128
#define B_BATCH   2
#define CAND_CAP  8192
#define N_LAYERS  5

typedef __attribute__((ext_vector_type(2))) float    v2f;
typedef __attribute__((ext_vector_type(8))) float    v8f;
typedef __attribute__((ext_vector_type(4))) unsigned v4u;
typedef __attribute__((ext_vector_type(8))) unsigned v8u;

// Native fp32 WMMA: D(16x16) = A(16x4) * B(4x16) + C
__device__ __forceinline__ v8f wmma_f32(v2f a, v2f b, v8f c) {
  return __builtin_amdgcn_wmma_f32_16x16x4_f32(
      /*neg_a=*/false, a, /*neg_b=*/false, b,
      /*c_mod=*/(short)0, c, /*reuse_a=*/false, /*reuse_b=*/false);
}

__device__ __forceinline__ v8f v8zero() {
  v8f z = {0.f,0.f,0.f,0.f,0.f,0.f,0.f,0.f};
  return z;
}

// ---- gfx1250 async global->LDS copy (ASYNCcnt) ----------------------------
__device__ __forceinline__ void async_ld_b128(void* lds, const void* g) {
  unsigned l = (unsigned)(uintptr_t)lds;   // low 32 bits of flat = LDS byte addr
  asm volatile("global_load_async_to_lds_b128 %0, %1, off"
               :: "v"(l), "v"(g) : "memory");
}
__device__ __forceinline__ void wait_async0() {
  asm volatile("s_wait_asynccnt 0x0" ::: "memory");
}

// ---- gfx1250 Tensor Data Mover: 2D tile load (TENSORcnt) ------------------
// D# group0: count=1 | lds_addr | global_addr(57b) | type=2
// D# group1: cfg(data_size/pad) | tensor dims | tile dims | dim0 stride
__device__ __forceinline__ void tdm_load_2d(void* lds, const void* gaddr,
                                            unsigned tile_d0, unsigned tile_d1,
                                            unsigned tensor_d0, unsigned tensor_d1,
                                            unsigned long long stride0_elems,
                                            unsigned cfg) {
  unsigned long long ga = (unsigned long long)(uintptr_t)gaddr;
  v4u g0;
  g0[0] = 1u;                                      // count=1 (valid descriptor)
  g0[1] = (unsigned)(uintptr_t)lds;                // lds_addr (bytes)
  g0[2] = (unsigned)ga;                            // global_addr[31:0]
  g0[3] = (unsigned)((ga >> 32) & 0x01FFFFFFu)     // global_addr[56:32]
        | 0x80000000u;                             // type=2 ("image")
  v8u g1;
  g1[0] = cfg;                                     // mask=0|data_size|pad fields
  g1[1] = (tensor_d0 & 0xFFFFu) << 16;             // tensor_dim0[15:0] @bit48
  g1[2] = (tensor_d0 >> 16) | ((tensor_d1 & 0xFFFFu) << 16);
  g1[3] = (tensor_d1 >> 16) | (tile_d0 << 16);     // tile_dim0 @bit112
  g1[4] = tile_d1 & 0xFFFFu;                       // tile_dim1; tile_dim2=0
  g1[5] = (unsigned)stride0_elems;                 // tensor_dim0_stride[31:0]
  g1[6] = (unsigned)((stride0_elems >> 32) & 0xFFFFu); // stride[47:32]; d1s=0
  g1[7] = 0u;
  asm volatile("tensor_load_to_lds %0, %1\n\t"
               "s_wait_tensorcnt 0x0"
               :: "s"(g0), "s"(g1) : "memory");
}

// data_size=4B (2<<16), pad_enable (1<<20), pad every 64 DWORDs (5<<22),
// pad amount 16 DWORDs (15<<25)  => LDS row stride 80 floats.
#define TDM_CFG_F32_PAD64_16 ((2u<<16) | (1u<<20) | (5u<<22) | (15u<<25))

// ---------------------------------------------------------------------------
// Weight pre-transposes so WMMA A-fragment loads are lane-coalesced.
// dw [5][cout][cin][3] -> wT [5][3][cin][cout]
__global__ __launch_bounds__(256) void k_wt(const float* __restrict__ dw,
                                            float* __restrict__ wT) {
  int i = blockIdx.x * 256 + threadIdx.x;
  if (i >= N_LAYERS * 3 * MD * MD) return;
  int co = i % MD; int r = i / MD;
  int ci = r % MD; r /= MD;
  int j  = r % 3;  int l = r / 3;
  wT[i] = dw[(((size_t)l * MD + co) * MD + ci) * 3 + j];
}

// w_atoms [a][k] -> waT [k][a]
__global__ __launch_bounds__(256) void k_wa(const float* __restrict__ wa,
                                            float* __restrict__ waT) {
  int i = blockIdx.x * 256 + threadIdx.x;
  if (i >= N_ATOMS * MD) return;
  int a = i % N_ATOMS; int k = i / N_ATOMS;
  waT[i] = wa[(size_t)a * MD + k];
}

__global__ __launch_bounds__(256) void k_zero(float* __restrict__ p, int n) {
  int i = blockIdx.x * 256 + threadIdx.x;
  if (i < n) p[i] = 0.0f;
}

// ---------------------------------------------------------------------------
// Stage 1: analysis filter bank. out[b,c,t] = sum_k bank[c,k]*orig[b,t+k-256]
// then relu, 20*log(x+1e-8).  WMMA: M=channel, N=time, K=tap.
// Weight K-chunks streamed via async global->LDS b128 copies.
__global__ __launch_bounds__(256) void k_filterbank(const float* __restrict__ orig,
                                                    const float* __restrict__ bank,
                                                    float* __restrict__ xout) {
  __shared__ float shO[575];        // orig window [t0-256, t0+319)
  __shared__ float shW[128 * 68];   // K-chunk; stride 68 (16B-aligned rows,
                                    // conflict-free A-frag column reads)
  const int b   = blockIdx.y;
  const int t0  = blockIdx.x * 64;
  const int tid = threadIdx.x;
  const int lane = tid & 31, wv = tid >> 5;
  const int half = lane >> 4, lm = lane & 15;
  const float* ob = orig + (size_t)b * N_SAMPLES;

  for (int i = tid; i < 575; i += 256) {
    int t = t0 - 256 + i;
    shO[i] = (t >= 0 && t < N_SAMPLES) ? ob[t] : 0.0f;
  }

  v8f acc[4];
  #pragma unroll
  for (int nt = 0; nt < 4; ++nt) acc[nt] = v8zero();

  const int arow = (wv * 16 + lm) * 68;
  for (int kc = 0; kc < 512; kc += 64) {
    __syncthreads();   // previous chunk fully consumed
    // async-copy 128 rows x 64 taps (16B chunks): 2048 chunks over 256 threads
    for (int i = tid; i < 128 * 16; i += 256) {
      int c = i >> 4, ch = i & 15;
      async_ld_b128(&shW[c * 68 + ch * 4], bank + (size_t)c * 512 + kc + ch * 4);
    }
    wait_async0();
    __syncthreads();
    #pragma unroll 4
    for (int kk = 0; kk < 64; kk += 4) {
      v2f a;
      a[0] = shW[arow + kk + 2 * half + 0];
      a[1] = shW[arow + kk + 2 * half + 1];
      const int kg = kc + kk + 2 * half;  // global tap index for this lane-half
      #pragma unroll
      for (int nt = 0; nt < 4; ++nt) {
        int n = nt * 16 + lm;
        v2f bb;
        bb[0] = shO[n + kg + 0];
        bb[1] = shO[n + kg + 1];
        acc[nt] = wmma_f32(a, bb, acc[nt]);
      }
    }
  }

  float* xb = xout + (size_t)b * MD * N_SAMPLES;
  #pragma unroll
  for (int nt = 0; nt < 4; ++nt) {
    int t = t0 + nt * 16 + lm;
    #pragma unroll
    for (int v = 0; v < 8; ++v) {
      int c = wv * 16 + v + 8 * half;
      float val = fmaxf(acc[nt][v], 0.0f);
      val = 20.0f * logf(val + 1e-8f);
      xb[(size_t)c * N_SAMPLES + t] = val;
    }
  }
}

// ---------------------------------------------------------------------------
// Stage 2: one dilated residual layer.
// y[co,t] = b[co] + sum_{j,ci} W[co,ci,j] * x[ci, t+(j-1)d]; x=leakyrelu(x+y,.2)
// Halo needs zero-fill at sequence ends -> keep predicated vector staging.
__global__ __launch_bounds__(256) void k_dilated(const float* __restrict__ xin,
                                                 float* __restrict__ xout,
                                                 const float* __restrict__ wTl, // [3][ci][co]
                                                 const float* __restrict__ bias,
                                                 int d) {
  __shared__ float shX[MD * 119];   // [ci][64 + 2d], stride W+1 (odd)
  const int b   = blockIdx.y;
  const int t0  = blockIdx.x * 64;
  const int tid = threadIdx.x;
  const int lane = tid & 31, wv = tid >> 5;
  const int half = lane >> 4, lm = lane & 15;
  const int W  = 64 + 2 * d;
  const int SW = W + 1;
  const float* xb = xin + (size_t)b * MD * N_SAMPLES;

  for (int i = tid; i < MD * W; i += 256) {
    int ci = i / W, c = i % W;
    int t = t0 - d + c;
    shX[ci * SW + c] = (t >= 0 && t < N_SAMPLES) ? xb[(size_t)ci * N_SAMPLES + t]
                                                 : 0.0f;
  }
  __syncthreads();

  v8f acc[4];
  #pragma unroll
  for (int nt = 0; nt < 4; ++nt) acc[nt] = v8zero();

  const int co = wv * 16 + lm;
  for (int j = 0; j < 3; ++j) {
    const int joff = j * d;
    const float* wj = wTl + (size_t)j * MD * MD;  // [ci][co]
    #pragma unroll 4
    for (int kk = 0; kk < MD; kk += 4) {
      const int k0 = kk + 2 * half;
      __builtin_prefetch(wj + (size_t)(k0 + 8) * MD + co, 0, 0);
      v2f a;
      a[0] = wj[(size_t)(k0 + 0) * MD + co];
      a[1] = wj[(size_t)(k0 + 1) * MD + co];
      #pragma unroll
      for (int nt = 0; nt < 4; ++nt) {
        int n = nt * 16 + lm + joff;
        v2f bb;
        bb[0] = shX[(k0 + 0) * SW + n];
        bb[1] = shX[(k0 + 1) * SW + n];
        acc[nt] = wmma_f32(a, bb, acc[nt]);
      }
    }
  }

  float* xob = xout + (size_t)b * MD * N_SAMPLES;
  #pragma unroll
  for (int nt = 0; nt < 4; ++nt) {
    int tl = nt * 16 + lm;
    #pragma unroll
    for (int v = 0; v < 8; ++v) {
      int c = wv * 16 + v + 8 * half;
      float y  = acc[nt][v] + bias[c];
      float xv = shX[c * SW + tl + d];
      float z  = xv + y;
      z = (z >= 0.0f) ? z : 0.2f * z;
      xob[(size_t)c * N_SAMPLES + t0 + tl] = z;
    }
  }
}

// ---------------------------------------------------------------------------
// Stage 3: atom logits GEMM fused with histogram (mode 0) or candidate
// collection above threshold (mode 1). feat never hits memory (saves 536 MB).
// x tile (128x64 f32, LDS stride 80 via TDM pad) loaded by the Tensor DMA.
__global__ __launch_bounds__(256) void k_feat(const float* __restrict__ x,
                                              const float* __restrict__ waT, // [k][a]
                                              const float* __restrict__ b_atoms,
                                              unsigned* __restrict__ hist,   // [B][1024]
                                              const unsigned* __restrict__ thr,
                                              float* __restrict__ candV,
                                              unsigned* __restrict__ candI,
                                              int* __restrict__ candCnt,
                                              int mode) {
  __shared__ float shX[MD * 80];    // [k][64], stride 80 -> conflict-free B reads
  __shared__ unsigned shH[1024];
  const int b   = blockIdx.z;
  const int a0  = blockIdx.y * 128;
  const int t0  = blockIdx.x * 64;
  const int tid = threadIdx.x;
  const int lane = tid & 31, wv = tid >> 5;
  const int half = lane >> 4, lm = lane & 15;

  if (mode == 0)
    for (int i = tid; i < 1024; i += 256) shH[i] = 0u;

  const float* xb = x + (size_t)b * MD * N_SAMPLES;
  if (wv == 0) {
    // one TDM descriptor moves the whole 32 KB tile, padding 16 DWORDs per
    // 64-DWORD row to realize the stride-80 LDS layout.
    tdm_load_2d(shX, xb + t0, /*tile*/64, MD, /*tensor*/N_SAMPLES, MD,
                (unsigned long long)N_SAMPLES, TDM_CFG_F32_PAD64_16);
  }
  __syncthreads();

  v8f acc[4];
  #pragma unroll
  for (int nt = 0; nt < 4; ++nt) acc[nt] = v8zero();

  const int am = a0 + wv * 16 + lm;
  #pragma unroll 4
  for (int kk = 0; kk < MD; kk += 4) {
    const int k0 = kk + 2 * half;
    __builtin_prefetch(waT + (size_t)(k0 + 8) * N_ATOMS + am, 0, 0);
    v2f a;
    a[0] = waT[(size_t)(k0 + 0) * N_ATOMS + am];
    a[1] = waT[(size_t)(k0 + 1) * N_ATOMS + am];
    #pragma unroll
    for (int nt = 0; nt < 4; ++nt) {
      int n = nt * 16 + lm;
      v2f bb;
      bb[0] = shX[(k0 + 0) * 80 + n];
      bb[1] = shX[(k0 + 1) * 80 + n];
      acc[nt] = wmma_f32(a, bb, acc[nt]);
    }
  }

  const unsigned tb = (mode == 1) ? thr[b] : 0u;
  #pragma unroll
  for (int nt = 0; nt < 4; ++nt) {
    int t = t0 + nt * 16 + lm;
    #pragma unroll
    for (int v = 0; v < 8; ++v) {
      int atom = a0 + wv * 16 + v + 8 * half;
      float val = acc[nt][v] + b_atoms[atom];
      unsigned u = __float_as_uint(val);
      unsigned key = (u & 0x80000000u) ? ~u : (u | 0x80000000u);
      if (mode == 0) {
        atomicAdd(&shH[key >> 22], 1u);
      } else if (key >= tb) {
        int pos = atomicAdd(&candCnt[b], 1);
        if (pos < CAND_CAP) {
          candV[b * CAND_CAP + pos] = val;
          candI[b * CAND_CAP + pos] = (unsigned)atom * N_SAMPLES + (unsigned)t;
        }
      }
    }
  }
  if (mode == 0) {
    __syncthreads();
    for (int i = tid; i < 1024; i += 256)
      if (shH[i]) atomicAdd(&hist[b * 1024 + i], shH[i]);
  }
}

// Find per-batch key-threshold bin whose suffix count >= K_SPARSE.
__global__ void k_threshold(const unsigned* __restrict__ hist,
                            unsigned* __restrict__ thr) {
  int b = threadIdx.x;
  if (b >= B_BATCH) return;
  unsigned cum = 0, t = 0;
  for (int bin = 1023; bin >= 0; --bin) {
    cum += hist[b * 1024 + bin];
    if (cum >= K_SPARSE) { t = (unsigned)bin << 22; break; }
  }
  thr[b] = t;
}

// Exact top-64 from the candidate pool (iterative argmax-extract).
__global__ __launch_bounds__(256) void k_topk(float* __restrict__ candV,
                                              unsigned* __restrict__ candI,
                                              const int* __restrict__ candCnt,
                                              float* __restrict__ topv,
                                              unsigned* __restrict__ topi) {
  __shared__ float sv[256];
  __shared__ int   si[256];
  const int b = blockIdx.x, tid = threadIdx.x;
  int n = candCnt[b]; if (n > CAND_CAP) n = CAND_CAP;
  float*    cv = candV + (size_t)b * CAND_CAP;
  unsigned* ci = candI + (size_t)b * CAND_CAP;
  for (int k = 0; k < K_SPARSE; ++k) {
    float best = -__builtin_inff(); int bi = -1;
    for (int i = tid; i < n; i += 256) {
      float v = cv[i];
      if (v > best || (v == best && bi >= 0 && ci[i] < ci[bi])) { best = v; bi = i; }
    }
    sv[tid] = best; si[tid] = bi;
    __syncthreads();
    for (int s = 128; s > 0; s >>= 1) {
      if (tid < s && sv[tid + s] > sv[tid]) { sv[tid] = sv[tid + s]; si[tid] = si[tid + s]; }
      __syncthreads();
    }
    if (tid == 0) {
      int sel = si[0];
      if (sel >= 0) {
        topv[b * K_SPARSE + k] = sv[0];
        topi[b * K_SPARSE + k] = ci[sel];
        cv[sel] = -__builtin_inff();
      } else {
        topv[b * K_SPARSE + k] = 0.0f;
        topi[b * K_SPARSE + k] = 0u;
      }
    }
    __syncthreads();
  }
}

// Stage 4: scatter-add atoms into the single shared buffer (as reference).
__global__ __launch_bounds__(256) void k_scatter(const float* __restrict__ atoms,
                                                 const float* __restrict__ topv,
                                                 const unsigned* __restrict__ topi,
                                                 float* __restrict__ buf) {
  const int sel = blockIdx.x;             // 0 .. B*K-1
  const float v = topv[sel];
  const unsigned idx = topi[sel];
  const int a = (int)(idx / N_SAMPLES);
  const int t = (int)(idx % N_SAMPLES);
  const float* ap = atoms + (size_t)a * ATOM_SIZE;
  for (int i = threadIdx.x; i < ATOM_SIZE; i += 256)
    atomicAdd(&buf[t + i], ap[i] * v);
}

__global__ __launch_bounds__(256) void k_out(const float* __restrict__ buf,
                                             float* __restrict__ out) {
  int t = blockIdx.x * 256 + threadIdx.x;
  if (t < N_SAMPLES) {
    float v = buf[t];
    out[t] = v;                 // batch 0
    out[N_SAMPLES + t] = v;     // batch 1 (shared buffer broadcast)
  }
}

// ---------------------------------------------------------------------------
extern "C" void kernel_launch(void* const* d_in, const int* in_sizes, int n_in,
                              void* d_out, int out_size, void* d_ws, size_t ws_size,
                              hipStream_t stream) {
  const float* orig    = (const float*)d_in[0];
  const float* bank    = (const float*)d_in[1];
  const float* dw      = (const float*)d_in[2];
  const float* db      = (const float*)d_in[3];
  const float* w_atoms = (const float*)d_in[4];
  const float* b_atoms = (const float*)d_in[5];
  const float* atoms   = (const float*)d_in[6];
  float* out = (float*)d_out;

  char* ws = (char*)d_ws;
  const size_t SZX = (size_t)B_BATCH * MD * N_SAMPLES * sizeof(float); // 33.5 MB
  float*    XA      = (float*)(ws);
  float*    XB      = (float*)(ws + SZX);
  char* p = ws + 2 * SZX;
  unsigned* hist    = (unsigned*)p;  p += B_BATCH * 1024 * 4;          // 8192
  int*      candCnt = (int*)p;       p += 256;
  unsigned* thr     = (unsigned*)p;  p += 256;
  float*    topv    = (float*)p;     p += 256;                          // 128 used
  unsigned* topi    = (unsigned*)p;  p += 256;
  float*    candV   = (float*)p;     p += (size_t)B_BATCH * CAND_CAP * 4;
  unsigned* candI   = (unsigned*)p;  p += (size_t)B_BATCH * CAND_CAP * 4;
  float*    buf     = (float*)p;     p += (size_t)2 * N_SAMPLES * 4;
  float*    wT      = (float*)p;     p += (size_t)N_LAYERS * 3 * MD * MD * 4;
  float*    waT     = (float*)p;     p += (size_t)N_ATOMS * MD * 4;

  // Per-call re-init (determinism): hist+candCnt+thr+topv+topi block, and buf.
  k_zero<<<dim3(10), 256, 0, stream>>>((float*)hist, 2048 + 64 + 64 + 64 + 64);
  k_zero<<<dim3((2 * N_SAMPLES + 255) / 256), 256, 0, stream>>>(buf, 2 * N_SAMPLES);

  // Weight transposes for coalesced WMMA A-fragment loads.
  k_wt<<<dim3((N_LAYERS * 3 * MD * MD + 255) / 256), 256, 0, stream>>>(dw, wT);
  k_wa<<<dim3((N_ATOMS * MD + 255) / 256), 256, 0, stream>>>(w_atoms, waT);

  const dim3 gconv(N_SAMPLES / 64, B_BATCH);
  k_filterbank<<<gconv, 256, 0, stream>>>(orig, bank, XA);

  const int dil[N_LAYERS] = {1, 3, 9, 27, 1};
  const float* src = XA; float* dst = XB;
  for (int i = 0; i < N_LAYERS; ++i) {
    k_dilated<<<gconv, 256, 0, stream>>>(src, dst,
                                         wT + (size_t)i * 3 * MD * MD,
                                         db + i * MD, dil[i]);
    float* tmp = dst; dst = (float*)src; src = tmp;
  }
  // final features are in `src`

  const dim3 gf(N_SAMPLES / 64, N_ATOMS / 128, B_BATCH);
  k_feat<<<gf, 256, 0, stream>>>(src, waT, b_atoms, hist, thr,
                                 candV, candI, candCnt, /*mode=*/0);
  k_threshold<<<1, 32, 0, stream>>>(hist, thr);
  k_feat<<<gf, 256, 0, stream>>>(src, waT, b_atoms, hist, thr,
                                 candV, candI, candCnt, /*mode=*/1);
  k_topk<<<B_BATCH, 256, 0, stream>>>(candV, candI, candCnt, topv, topi);
  k_scatter<<<B_BATCH * K_SPARSE, 256, 0, stream>>>(atoms, topv, topi, buf);
  k_out<<<(N_SAMPLES + 255) / 256, 256, 0, stream>>>(buf, out);
}